// QuantumRegression__gen033_65481071398114
// MI455X (gfx1250) — compile-verified
//
#include <hip/hip_runtime.h>

// Quantum state-vector regression, MI455X (gfx1250).
// One workgroup per batch state: the whole 128KB state lives in the 320KB WGP
// LDS (CDNA5-unique capacity). State loaded via async global->LDS copies
// (ASYNCcnt), overlapped with staging the 42 gate matrices into LDS.
// Register blocking: each thread owns a 4-bit subcube per pass, so a full
// layer (14 gates + 14 CNOTs) needs only 5 LDS sweeps. XOR bank swizzle keeps
// all pass strides (near) conflict-free on the 64 LDS banks (wave32).

#define NW   14
#define DSZ  (1 << NW)        // 16384 amplitudes
#define NL   3
#define TPB  1024
#define EPT  (DSZ / TPB)      // 16 elements / thread
#define NGF  (NL * NW * 8)    // 336 gate coefficient floats

// ---------------------------------------------------------------------------
// Precompute the 42 Rot matrices (RZ*RY*RX) -> 8 floats each in d_ws.
// ---------------------------------------------------------------------------
extern "C" __global__ void qr_prep_gates(const float* __restrict__ var_params,
                                         float* __restrict__ gbuf) {
  int g = threadIdx.x;
  if (g >= NL * NW) return;
  float tx = 0.5f * var_params[g * 3 + 0];
  float ty = 0.5f * var_params[g * 3 + 1];
  float tz = 0.5f * var_params[g * 3 + 2];
  float sx, cx, sy, cy, sz, cz;
  sincosf(tx, &sx, &cx);
  sincosf(ty, &sy, &cy);
  sincosf(tz, &sz, &cz);
  float a = cy * cx, b = sy * sx, c = sy * cx, d = cy * sx;
  float* o = gbuf + g * 8;
  o[0] =  cz * a + sz * b;   // U00 re
  o[1] =  cz * b - sz * a;   // U00 im
  o[2] = -cz * c - sz * d;   // U01 re
  o[3] =  sz * c - cz * d;   // U01 im
  o[4] =  cz * c + sz * d;   // U10 re  (= -conj(U01))
  o[5] =  sz * c - cz * d;   // U10 im
  o[6] =  cz * a + sz * b;   // U11 re  (=  conj(U00))
  o[7] =  sz * a - cz * b;   // U11 im
}

// ---- helpers ---------------------------------------------------------------
__device__ __forceinline__ unsigned swz(unsigned i) {
  // bank swizzle: XOR bits 5..2 with bits 9..6; constant within 64-word blocks
  return i ^ (((i >> 6) & 0xFu) << 2);
}

// apply 2x2 complex gate on local bit BL of the thread's 16-element subcube
// gm points into LDS (32B aligned) -> two ds_load_b128 broadcasts
template <int BL>
__device__ __forceinline__ void gate16(float* xr, float* xi, const float* gm) {
  const float4 gA = *(const float4*)(gm);
  const float4 gB = *(const float4*)(gm + 4);
  const float u00r = gA.x, u00i = gA.y, u01r = gA.z, u01i = gA.w;
  const float u10r = gB.x, u10i = gB.y, u11r = gB.z, u11i = gB.w;
#pragma unroll
  for (int j0 = 0; j0 < 16; ++j0) {
    if (j0 & (1 << BL)) continue;
    const int j1 = j0 | (1 << BL);
    const float ar = xr[j0], ai = xi[j0], br = xr[j1], bi = xi[j1];
    xr[j0] = u00r * ar - u00i * ai + u01r * br - u01i * bi;
    xi[j0] = u00r * ai + u00i * ar + u01r * bi + u01i * br;
    xr[j1] = u10r * ar - u10i * ai + u11r * br - u11i * bi;
    xi[j1] = u10r * ai + u10i * ar + u11r * bi + u11i * br;
  }
}

// CNOT: control local bit BC, target local bit BT (pure register permute)
template <int BC, int BT>
__device__ __forceinline__ void cx16(float* xr, float* xi) {
#pragma unroll
  for (int j = 0; j < 16; ++j) {
    if ((j & (1 << BC)) && !(j & (1 << BT))) {
      const int k = j | (1 << BT);
      const float tr = xr[j], ti = xi[j];
      xr[j] = xr[k]; xi[j] = xi[k];
      xr[k] = tr;    xi[k] = ti;
    }
  }
}

#define PASS_BEGIN(IDXEXPR)                                                    \
  {                                                                            \
    unsigned a_[16]; float xr[16], xi[16];                                     \
    _Pragma("unroll") for (int j = 0; j < 16; ++j) {                           \
      unsigned i_ = (IDXEXPR); a_[j] = swz(i_);                                \
    }                                                                          \
    _Pragma("unroll") for (int j = 0; j < 16; ++j) {                           \
      xr[j] = R[a_[j]]; xi[j] = I[a_[j]];                                      \
    }

#define PASS_END()                                                             \
    _Pragma("unroll") for (int j = 0; j < 16; ++j) {                           \
      R[a_[j]] = xr[j]; I[a_[j]] = xi[j];                                      \
    }                                                                          \
    __syncthreads();                                                           \
  }

// ---------------------------------------------------------------------------
// Main kernel: one block (32 wave32s) per batch element.
// ---------------------------------------------------------------------------
extern "C" __global__ __launch_bounds__(TPB)
void qr_main(const float* __restrict__ sr, const float* __restrict__ si,
             const float* __restrict__ gates,
             const float* __restrict__ head_w, const float* __restrict__ head_b,
             float* __restrict__ out) {
  extern __shared__ float smem[];
  float* R  = smem;                 // 16384 floats (swizzled layout)
  float* I  = smem + DSZ;           // 16384 floats (swizzled layout)
  float* F  = smem + 2 * DSZ;       // 14 feature accumulators (pad to 16)
  float* GL = smem + 2 * DSZ + 16;  // 336 staged gate coefficients

  const unsigned T = threadIdx.x;
  const int b      = blockIdx.x;
  const int lane   = T & 31;

  // ---- CDNA5 async copy HBM -> LDS (16B per lane per op, ASYNCcnt-tracked)
  {
    unsigned long long baseR = (unsigned long long)(sr + (size_t)b * DSZ);
    unsigned long long baseI = (unsigned long long)(si + (size_t)b * DSZ);
#pragma unroll
    for (int k = 0; k < EPT / 4; ++k) {
      unsigned c    = T + (unsigned)k * TPB;      // 16B chunk id
      unsigned goff = c * 16u;                    // global byte offset (GVS)
      unsigned ldsR = (unsigned)(unsigned long long)(&R[swz(c * 4u)]);
      unsigned ldsI = (unsigned)(unsigned long long)(&I[swz(c * 4u)]);
      asm volatile("global_load_async_to_lds_b128 %0, %1, %2"
                   :: "v"(ldsR), "v"(goff), "s"(baseR) : "memory");
      asm volatile("global_load_async_to_lds_b128 %0, %1, %2"
                   :: "v"(ldsI), "v"(goff), "s"(baseI) : "memory");
    }
  }
  // ---- overlap with the DMA: stage gate matrices + init feature accums
  if (T < NW) F[T] = 0.0f;
  if (T < NGF) GL[T] = gates[T];
  asm volatile("s_wait_asynccnt 0" ::: "memory");
  __syncthreads();

  // ---- 3 layers, 5 register-blocked LDS passes each --------------------
  for (int l = 0; l < NL; ++l) {
    const float* G = GL + l * (NW * 8);

    // Pass 1: local bits 13..10 = wires 0..3 (wire w <-> bit 13-w)
    PASS_BEGIN(((unsigned)j << 10) | T)
      gate16<3>(xr, xi, G + 0 * 8);   // wire 0
      gate16<2>(xr, xi, G + 1 * 8);   // wire 1
      gate16<1>(xr, xi, G + 2 * 8);   // wire 2
      gate16<0>(xr, xi, G + 3 * 8);   // wire 3
      cx16<3, 2>(xr, xi);             // CX(0,1)
      cx16<2, 1>(xr, xi);             // CX(1,2)
      cx16<1, 0>(xr, xi);             // CX(2,3)
    PASS_END()

    // Pass 2: local bits 10..7 = wires 3..6
    PASS_BEGIN(((T >> 7) << 11) | ((unsigned)j << 7) | (T & 127u))
      gate16<2>(xr, xi, G + 4 * 8);   // wire 4
      gate16<1>(xr, xi, G + 5 * 8);   // wire 5
      gate16<0>(xr, xi, G + 6 * 8);   // wire 6
      cx16<3, 2>(xr, xi);             // CX(3,4)
      cx16<2, 1>(xr, xi);             // CX(4,5)
      cx16<1, 0>(xr, xi);             // CX(5,6)
    PASS_END()

    // Pass 3: local bits 7..4 = wires 6..9
    PASS_BEGIN(((T >> 4) << 8) | ((unsigned)j << 4) | (T & 15u))
      gate16<2>(xr, xi, G + 7 * 8);   // wire 7
      gate16<1>(xr, xi, G + 8 * 8);   // wire 8
      gate16<0>(xr, xi, G + 9 * 8);   // wire 9
      cx16<3, 2>(xr, xi);             // CX(6,7)
      cx16<2, 1>(xr, xi);             // CX(7,8)
      cx16<1, 0>(xr, xi);             // CX(8,9)
    PASS_END()

    // Pass 4: local bits 4..1 = wires 9..12
    PASS_BEGIN(((T >> 1) << 5) | ((unsigned)j << 1) | (T & 1u))
      gate16<2>(xr, xi, G + 10 * 8);  // wire 10
      gate16<1>(xr, xi, G + 11 * 8);  // wire 11
      gate16<0>(xr, xi, G + 12 * 8);  // wire 12
      cx16<3, 2>(xr, xi);             // CX(9,10)
      cx16<2, 1>(xr, xi);             // CX(10,11)
      cx16<1, 0>(xr, xi);             // CX(11,12)
    PASS_END()

    // Pass 5: j3->bit13 (wire 0), j2->bit2 (w11), j1->bit1 (w12), j0->bit0 (w13)
    PASS_BEGIN((((unsigned)j & 8u) << 10) | (T << 3) | ((unsigned)j & 7u))
      gate16<0>(xr, xi, G + 13 * 8);  // wire 13
      cx16<1, 0>(xr, xi);             // CX(12,13)
      cx16<0, 3>(xr, xi);             // CX(13,0)
    PASS_END()
  }

  // ---- <Z_w> features: signed |psi|^2 accumulation ----------------------
  float acc[NW];
#pragma unroll
  for (int w = 0; w < NW; ++w) acc[w] = 0.0f;
#pragma unroll
  for (int k = 0; k < EPT; ++k) {
    unsigned idx = T + (unsigned)k * TPB;      // logical index
    unsigned a   = swz(idx);
    float pr = R[a], pi = I[a];
    float pv = pr * pr + pi * pi;
#pragma unroll
    for (int w = 0; w < NW; ++w)
      acc[w] += ((idx >> (NW - 1 - w)) & 1u) ? -pv : pv;
  }
  // wave32 butterfly reduction, then one ds_add_f32 per wave per feature
#pragma unroll
  for (int off = 16; off > 0; off >>= 1) {
#pragma unroll
    for (int w = 0; w < NW; ++w)
      acc[w] += __shfl_xor(acc[w], off, 32);
  }
  if (lane == 0) {
#pragma unroll
    for (int w = 0; w < NW; ++w) atomicAdd(&F[w], acc[w]);
  }
  __syncthreads();

  // ---- linear head ------------------------------------------------------
  if (T == 0) {
    float o = head_b[0];
#pragma unroll
    for (int w = 0; w < NW; ++w) o += F[w] * head_w[w];
    out[b] = o;
  }
}

// ---------------------------------------------------------------------------
extern "C" void kernel_launch(void* const* d_in, const int* in_sizes, int n_in,
                              void* d_out, int out_size, void* d_ws, size_t ws_size,
                              hipStream_t stream) {
  const float* sr = (const float*)d_in[0];
  const float* si = (const float*)d_in[1];
  const float* vp = (const float*)d_in[2];
  const float* hw = (const float*)d_in[3];
  const float* hb = (const float*)d_in[4];
  float* outp = (float*)d_out;
  float* gbuf = (float*)d_ws;                  // 42*8 floats = 1344 B

  const int B = in_sizes[0] / DSZ;             // 1024

  qr_prep_gates<<<1, 64, 0, stream>>>(vp, gbuf);

  const size_t shbytes = (size_t)(2 * DSZ + 16 + NGF) * sizeof(float);
  qr_main<<<B, TPB, shbytes, stream>>>(sr, si, gbuf, hw, hb, outp);
}